// LeroNet_50766513439352
// MI455X (gfx1250) — compile-verified
//
#include <hip/hip_runtime.h>

typedef float v2f __attribute__((ext_vector_type(2)));
typedef float v4f __attribute__((ext_vector_type(4)));
typedef float v8f __attribute__((ext_vector_type(8)));

#define N_ROWS  8184      // 8 trees * 1023 nodes
#define N_PAD   8192
#define DM      64        // D_MODEL
#define DI      128       // D_INNER
#define DS      64        // D_STATE
#define DBC_LD  144       // 132 padded to 144
#define HSZ     ((size_t)(DI*DS))

__device__ __forceinline__ float silu_f(float v) {
    return v / (1.0f + __expf(-v));
}
__device__ __forceinline__ float softplus_f(float v) {
    return (v > 20.0f) ? v : log1pf(__expf(v));
}

// One wave computes a 16x16 f32 tile of A[M x K] @ W[K x wcols] with
// V_WMMA_F32_16X16X4_F32.  Out-of-range rows/cols are handled by address
// clamping: clamped A rows only affect pad output rows (never consumed),
// clamped B cols only write into padding columns (never consumed).
template<int K>
__device__ __forceinline__ v8f wmma_tile(const float* __restrict__ A, int lda,
                                         const float* __restrict__ W, int ldw, int wcols,
                                         int row0, int col0, int valid_rows) {
    const int lane = threadIdx.x & 31;
    const int half = lane >> 4;           // K sub-pair: lanes 0-15 -> K0,K1 ; 16-31 -> K2,K3
    const int l16  = lane & 15;
    const int arow = min(row0 + l16, valid_rows - 1);
    const int bcol = min(col0 + l16, wcols - 1);
    const float* Ap = A + (size_t)arow * lda + 2*half;
    const float* Wp = W + (size_t)(2*half) * ldw + bcol;
    v8f acc = {0.f,0.f,0.f,0.f,0.f,0.f,0.f,0.f};
    #pragma unroll
    for (int k = 0; k < K; k += 4) {
        v2f a = *(const v2f*)(Ap + k);            // contiguous -> global_load_b64
        v2f b;
        b.x = Wp[(size_t)k * ldw];
        b.y = Wp[(size_t)(k + 1) * ldw];
        acc = __builtin_amdgcn_wmma_f32_16x16x4_f32(false, a, false, b,
                                                    (short)0, acc, false, false);
    }
    return acc;
}

__device__ __forceinline__ void tile_coords(int ntc, int* row0, int* col0) {
    const int tile = blockIdx.x * 8 + (threadIdx.x >> 5);   // 8 waves / block
    *row0 = (tile / ntc) * 16;
    *col0 = (tile % ntc) * 16;
}

// h0 = x @ Wf + bf
__global__ void gemm_h0(const float* __restrict__ x, const float* __restrict__ Wf,
                        const float* __restrict__ bf, float* __restrict__ h0) {
    int row0, col0; tile_coords(DM/16, &row0, &col0);
    v8f c = wmma_tile<DM>(x, DM, Wf, DM, DM, row0, col0, N_ROWS);
    const int lane = threadIdx.x & 31;
    const int col  = col0 + (lane & 15);
    const int rb   = row0 + 8*(lane >> 4);
    const float bias = bf[col];
    #pragma unroll
    for (int r = 0; r < 8; r++) h0[(size_t)(rb+r)*DM + col] = c[r] + bias;
}

// xz = h0 @ W_in + b_in ; xi = silu(xz[:, :128]) ; z = xz[:, 128:]
__global__ void gemm_xz(const float* __restrict__ h0, const float* __restrict__ W_in,
                        const float* __restrict__ b_in,
                        float* __restrict__ xi, float* __restrict__ z) {
    int row0, col0; tile_coords((2*DI)/16, &row0, &col0);
    v8f c = wmma_tile<DM>(h0, DM, W_in, 2*DI, 2*DI, row0, col0, N_PAD);
    const int lane = threadIdx.x & 31;
    const int col  = col0 + (lane & 15);
    const int rb   = row0 + 8*(lane >> 4);
    const float bias = b_in[col];
    if (col0 < DI) {
        #pragma unroll
        for (int r = 0; r < 8; r++)
            xi[(size_t)(rb+r)*DI + col] = silu_f(c[r] + bias);
    } else {
        #pragma unroll
        for (int r = 0; r < 8; r++)
            z[(size_t)(rb+r)*DI + (col - DI)] = c[r] + bias;
    }
}

// dbc = xi @ W_xp   (132 real cols, stored with stride 144; cols >=132 are junk pad)
__global__ void gemm_dbc(const float* __restrict__ xi, const float* __restrict__ W_xp,
                         float* __restrict__ dbc) {
    int row0, col0; tile_coords(9, &row0, &col0);
    v8f c = wmma_tile<DI>(xi, DI, W_xp, 132, 132, row0, col0, N_PAD);
    const int lane = threadIdx.x & 31;
    const int col  = col0 + (lane & 15);
    const int rb   = row0 + 8*(lane >> 4);
    #pragma unroll
    for (int r = 0; r < 8; r++) dbc[(size_t)(rb+r)*DBC_LD + col] = c[r];
}

// dt = softplus(dbc[:, :4] @ W_dt + b_dt)   (K=4, plain VALU)
__global__ void dt_kernel(const float* __restrict__ dbc, const float* __restrict__ W_dt,
                          const float* __restrict__ b_dt, float* __restrict__ dt) {
    const int tid = blockIdx.x * blockDim.x + threadIdx.x;
    const int row = tid >> 7;
    const int d   = tid & (DI-1);
    float s = b_dt[d];
    #pragma unroll
    for (int r = 0; r < 4; r++) s += dbc[(size_t)row*DBC_LD + r] * W_dt[r*DI + d];
    dt[(size_t)row*DI + d] = softplus_f(s);
}

// ---- Tree scan: one block (8 waves) per node; each wave owns 16 d_inner rows.
// Lane layout: lanes 0-15 -> row pair even, lanes 16-31 -> row pair odd;
// each lane covers a float4 of the 64-wide state dim  => b128 coalesced traffic.

__global__ void scan_leaf(const float* __restrict__ xi, const float* __restrict__ dt,
                          const float* __restrict__ dbc, float* __restrict__ hout) {
    const int n    = blockIdx.x;            // 0 .. 8*512-1
    const int tree = n >> 9;
    const int i    = n & 511;
    const int r    = tree*1023 + 511 + i;
    const int wave = threadIdx.x >> 5;
    const int lane = threadIdx.x & 31;
    const int half = lane >> 4;
    const int sq   = (lane & 15) * 4;
    float* hb = hout + (size_t)n * HSZ;
    #pragma unroll
    for (int it = 0; it < 8; it++) {
        const int d = wave*16 + it*2 + half;
        const float dtx = dt[(size_t)r*DI + d] * xi[(size_t)r*DI + d];
        const v4f B = *(const v4f*)&dbc[(size_t)r*DBC_LD + 4 + sq];
        v4f h;
        #pragma unroll
        for (int j = 0; j < 4; j++) h[j] = dtx * B[j];
        *(v4f*)&hb[d*DS + sq] = h;
    }
}

__global__ void scan_level(const float* __restrict__ xi, const float* __restrict__ dt,
                           const float* __restrict__ dbc, const float* __restrict__ A_log,
                           const float* __restrict__ hchild, float* __restrict__ hout,
                           int level) {
    const int p    = blockIdx.x;
    const int tree = p >> level;
    const int i    = p & ((1 << level) - 1);
    const int r    = tree*1023 + ((1 << level) - 1) + i;
    const int wave = threadIdx.x >> 5;
    const int lane = threadIdx.x & 31;
    const int half = lane >> 4;
    const int sq   = (lane & 15) * 4;
    const float* c0 = hchild + ((size_t)(tree << (level+1)) + 2*i) * HSZ;
    const float* c1 = c0 + HSZ;
    float* hb = hout + (size_t)p * HSZ;
    #pragma unroll
    for (int it = 0; it < 8; it++) {
        const int d = wave*16 + it*2 + half;
        const float dtv = dt[(size_t)r*DI + d];
        const float dtx = dtv * xi[(size_t)r*DI + d];
        const v4f al  = *(const v4f*)&A_log[d*DS + sq];
        const v4f B   = *(const v4f*)&dbc[(size_t)r*DBC_LD + 4 + sq];
        const v4f h0v = *(const v4f*)&c0[d*DS + sq];
        const v4f h1v = *(const v4f*)&c1[d*DS + sq];
        v4f h;
        #pragma unroll
        for (int j = 0; j < 4; j++) {
            const float dA = __expf(-dtv * __expf(al[j]));
            h[j] = dA * (h0v[j] + h1v[j]) + dtx * B[j];
        }
        *(v4f*)&hb[d*DS + sq] = h;
    }
}

// Root (d=0): compute h_root and immediately y = h.C + D_skip*xi (only roots matter)
__global__ void scan_root(const float* __restrict__ xi, const float* __restrict__ dt,
                          const float* __restrict__ dbc, const float* __restrict__ A_log,
                          const float* __restrict__ D_skip, const float* __restrict__ hchild,
                          float* __restrict__ yroot) {
    const int tree = blockIdx.x;
    const int r    = tree*1023;
    const int wave = threadIdx.x >> 5;
    const int lane = threadIdx.x & 31;
    const int half = lane >> 4;
    const int sq   = (lane & 15) * 4;
    const float* c0 = hchild + (size_t)(tree*2) * HSZ;
    const float* c1 = c0 + HSZ;
    #pragma unroll
    for (int it = 0; it < 8; it++) {
        const int d = wave*16 + it*2 + half;
        const float dtv = dt[(size_t)r*DI + d];
        const float dtx = dtv * xi[(size_t)r*DI + d];
        const v4f al  = *(const v4f*)&A_log[d*DS + sq];
        const v4f B   = *(const v4f*)&dbc[(size_t)r*DBC_LD + 4 + sq];
        const v4f C   = *(const v4f*)&dbc[(size_t)r*DBC_LD + 68 + sq];
        const v4f h0v = *(const v4f*)&c0[d*DS + sq];
        const v4f h1v = *(const v4f*)&c1[d*DS + sq];
        float part = 0.f;
        #pragma unroll
        for (int j = 0; j < 4; j++) {
            const float dA = __expf(-dtv * __expf(al[j]));
            const float h  = dA * (h0v[j] + h1v[j]) + dtx * B[j];
            part += h * C[j];
        }
        // reduce across the 16 lanes that share this row (xor offsets stay in-half)
        #pragma unroll
        for (int off = 8; off > 0; off >>= 1) part += __shfl_xor(part, off, 32);
        if ((lane & 15) == 0)
            yroot[tree*DI + d] = part + D_skip[d] * xi[(size_t)r*DI + d];
    }
}

// out = ((yroot * silu(z_root)) @ W_out + b_out) @ W_cost + b_cost  -> 8 scalars
__global__ void final_kernel(const float* __restrict__ yroot, const float* __restrict__ z,
                             const float* __restrict__ W_out, const float* __restrict__ b_out,
                             const float* __restrict__ W_cost, const float* __restrict__ b_cost,
                             float* __restrict__ out) {
    const int tree = threadIdx.x >> 5;
    const int lane = threadIdx.x & 31;
    const int r    = tree*1023;
    float cost = 0.f;
    #pragma unroll
    for (int cc = 0; cc < 2; cc++) {
        const int c = lane + 32*cc;
        float o = b_out[c];
        for (int d = 0; d < DI; d++) {
            const float yz = yroot[tree*DI + d] * silu_f(z[(size_t)r*DI + d]);
            o += yz * W_out[d*DM + c];
        }
        cost += o * W_cost[c];
    }
    #pragma unroll
    for (int off = 16; off > 0; off >>= 1) cost += __shfl_xor(cost, off, 32);
    if (lane == 0) out[tree] = cost + b_cost[0];
}

extern "C" void kernel_launch(void* const* d_in, const int* in_sizes, int n_in,
                              void* d_out, int out_size, void* d_ws, size_t ws_size,
                              hipStream_t stream) {
    const float* x      = (const float*)d_in[0];
    const float* Wf     = (const float*)d_in[1];
    const float* bf     = (const float*)d_in[2];
    const float* W_in   = (const float*)d_in[3];
    const float* b_in   = (const float*)d_in[4];
    const float* W_xp   = (const float*)d_in[5];
    const float* W_dt   = (const float*)d_in[6];
    const float* b_dt   = (const float*)d_in[7];
    const float* A_log  = (const float*)d_in[8];
    const float* D_skip = (const float*)d_in[9];
    const float* W_out  = (const float*)d_in[10];
    const float* b_out  = (const float*)d_in[11];
    const float* W_cost = (const float*)d_in[12];
    const float* b_cost = (const float*)d_in[13];

    float* ws    = (float*)d_ws;
    float* h0    = ws;                                  // 8192*64
    float* xi    = h0    + (size_t)N_PAD*DM;            // 8192*128
    float* z     = xi    + (size_t)N_PAD*DI;            // 8192*128
    float* dbc   = z     + (size_t)N_PAD*DI;            // 8192*144
    float* dtb   = dbc   + (size_t)N_PAD*DBC_LD;        // 8192*128
    float* yroot = dtb   + (size_t)N_PAD*DI;            // 8*128
    float* hA    = yroot + 8*DI;                        // 8*512*128*64
    float* hB    = hA    + (size_t)8*512*DI*DS;         // 8*256*128*64

    // 8 waves (tiles) per 256-thread block
    gemm_h0 <<<dim3((N_PAD/16)*(DM/16)/8),     dim3(256), 0, stream>>>(x, Wf, bf, h0);
    gemm_xz <<<dim3((N_PAD/16)*((2*DI)/16)/8), dim3(256), 0, stream>>>(h0, W_in, b_in, xi, z);
    gemm_dbc<<<dim3((N_PAD/16)*9/8),           dim3(256), 0, stream>>>(xi, W_xp, dbc);
    dt_kernel<<<dim3((N_PAD*DI)/256),          dim3(256), 0, stream>>>(dbc, W_dt, b_dt, dtb);

    scan_leaf<<<dim3(8*512), dim3(256), 0, stream>>>(xi, dtb, dbc, hA);
    for (int L = 8; L >= 1; --L) {
        float* child = (L & 1) ? hB : hA;   // level L+1 output
        float* outb  = (L & 1) ? hA : hB;
        scan_level<<<dim3(8 << L), dim3(256), 0, stream>>>(xi, dtb, dbc, A_log, child, outb, L);
    }
    scan_root  <<<dim3(8), dim3(256), 0, stream>>>(xi, dtb, dbc, A_log, D_skip, hA, yroot);
    final_kernel<<<dim3(1), dim3(256), 0, stream>>>(yroot, z, W_out, b_out, W_cost, b_cost,
                                                    (float*)d_out);
}